// Attention_80564996538935
// MI455X (gfx1250) — compile-verified
//
#include <hip/hip_runtime.h>

#define N_   4
#define NQ_  256
#define NV_  512
#define NE_  256
#define QT   16          // queries per block
#define PSTR 516         // padded logits/probs row stride (floats): conflict-free WMMA A reads
#define HSTR 260         // padded heads row stride

typedef float v2f __attribute__((ext_vector_type(2)));
typedef float v8f __attribute__((ext_vector_type(8)));

// CDNA5 hardware tanh (TRANS pipe, co-executes with VALU FMAs).
__device__ __forceinline__ float tanh_hw(float x) {
#if __has_builtin(__builtin_amdgcn_tanhf)
  return __builtin_amdgcn_tanhf(x);
#else
  float y;
  asm("v_tanh_f32 %0, %1" : "=v"(y) : "v"(x));
  return y;
#endif
}

__global__ __launch_bounds__(512) void attn_additive_fused(
    const float* __restrict__ query, const float* __restrict__ context,
    const float* __restrict__ memv,  const float* __restrict__ w_logit,
    const float* __restrict__ b_logit, const float* __restrict__ temp,
    const float* __restrict__ w_reduce, const float* __restrict__ b_reduce,
    float* __restrict__ out)
{
  __shared__ float probs_s[QT][PSTR];   // logits -> probs (33 KB)
  __shared__ float heads_s[QT][HSTR];   // phase1: staged query tile; phase3/4: heads (16.6 KB)
  __shared__ float w_s[NE_];            // w_logit (1 KB)

  const int n    = blockIdx.x >> 4;     // NQ/QT = 16 tiles per batch
  const int qt   = blockIdx.x & 15;
  const int q0   = qt * QT;
  const int tid  = threadIdx.x;         // 0..511  (16 wave32s)
  const int lane = tid & 31;
  const int wave = tid >> 5;            // 0..15

  const float* qbase = query + ((size_t)n * NQ_ + q0) * NE_;
  const float* cbase = context + (size_t)n * NV_ * NE_;
  const float* mbase = memv    + (size_t)n * NV_ * NE_;

  // ---- Stage query tile + w_logit into LDS (coalesced) ----
  for (int i = tid; i < QT * NE_; i += 512)
    heads_s[i >> 8][i & 255] = qbase[i];
  for (int i = tid; i < NE_; i += 512)
    w_s[i] = w_logit[i];
  __syncthreads();

  // ---- Phase 1: logits[q][v] = sum_e tanh(q[e]+c[v][e]) * w[e]  (TRANS-bound) ----
  {
    const int v = tid;                           // 512 threads == NV: one v each
    const float* crow = cbase + (size_t)v * NE_;
    float acc[QT];
#pragma unroll
    for (int q = 0; q < QT; ++q) acc[q] = 0.f;

    for (int e = 0; e < NE_; e += 4) {
      if ((e & 31) == 0) __builtin_prefetch(crow + e + 64, 0, 1);  // global_prefetch_b8
      float4 c4 = *(const float4*)(crow + e);
      float w0 = w_s[e], w1 = w_s[e + 1], w2 = w_s[e + 2], w3 = w_s[e + 3];
#pragma unroll
      for (int q = 0; q < QT; ++q) {
        acc[q] += tanh_hw(heads_s[q][e + 0] + c4.x) * w0;
        acc[q] += tanh_hw(heads_s[q][e + 1] + c4.y) * w1;
        acc[q] += tanh_hw(heads_s[q][e + 2] + c4.z) * w2;
        acc[q] += tanh_hw(heads_s[q][e + 3] + c4.w) * w3;
      }
    }
    const float bl   = b_logit[0];
    const float invt = 1.0f / temp[0];
#pragma unroll
    for (int q = 0; q < QT; ++q)
      probs_s[q][v] = (acc[q] + bl) * invt;
  }
  __syncthreads();

  // ---- Phase 2: softmax over v (one wave per q row) ----
  {
    const int q = wave;
    float m = -3.4e38f;
    for (int i = lane; i < NV_; i += 32) m = fmaxf(m, probs_s[q][i]);
#pragma unroll
    for (int off = 16; off; off >>= 1) m = fmaxf(m, __shfl_xor(m, off, 32));
    float s = 0.f;
    for (int i = lane; i < NV_; i += 32) {
      float ex = __expf(probs_s[q][i] - m);
      probs_s[q][i] = ex;
      s += ex;
    }
#pragma unroll
    for (int off = 16; off; off >>= 1) s += __shfl_xor(s, off, 32);
    const float inv = 1.0f / s;
    for (int i = lane; i < NV_; i += 32) probs_s[q][i] *= inv;
  }
  __syncthreads();

  // ---- Phase 3: heads = leaky_relu(probs @ memory), WMMA f32 16x16x4 ----
  // Wave `wave` owns the 16-wide e-tile e0 = wave*16. K loop over 512 v's, 4 at a time.
  {
    const int half = lane >> 4;          // A/B K-split across half-waves (ISA 32-bit layout)
    const int lh   = lane & 15;
    const int e0   = wave * 16;
    v8f c = {};
    for (int kk = 0; kk < NV_; kk += 4) {
      const int k = kk + 2 * half;
      v2f a, b;
      a.x = probs_s[lh][k];
      a.y = probs_s[lh][k + 1];
      b.x = mbase[(size_t)k * NE_ + e0 + lh];
      b.y = mbase[(size_t)(k + 1) * NE_ + e0 + lh];
      c = __builtin_amdgcn_wmma_f32_16x16x4_f32(false, a, false, b, (short)0, c,
                                                false, false);
    }
    // D layout: lane<16 -> N=lane, M=r ; lane>=16 -> N=lane-16, M=r+8
#pragma unroll
    for (int r = 0; r < 8; ++r) {
      float h = c[r];
      h = (h > 0.f) ? h : 0.01f * h;     // leaky_relu
      heads_s[r + 8 * half][e0 + lh] = h;
    }
  }
  __syncthreads();

  // ---- Phase 4: out = heads @ w_reduce^T + b_reduce, WMMA f32 16x16x4 ----
  {
    const int half = lane >> 4;
    const int lh   = lane & 15;
    const int o0   = wave * 16;
    v8f c = {};
    for (int kk = 0; kk < NE_; kk += 4) {
      const int k = kk + 2 * half;
      v2f a, b;
      a.x = heads_s[lh][k];
      a.y = heads_s[lh][k + 1];
      b.x = w_reduce[(size_t)(o0 + lh) * NE_ + k];      // B[k][o] = w_reduce[o][k]
      b.y = w_reduce[(size_t)(o0 + lh) * NE_ + k + 1];
      c = __builtin_amdgcn_wmma_f32_16x16x4_f32(false, a, false, b, (short)0, c,
                                                false, false);
    }
    const float bo = b_reduce[o0 + lh];
    float* obase = out + ((size_t)n * NQ_ + q0) * NE_;
#pragma unroll
    for (int r = 0; r < 8; ++r)
      obase[(size_t)(r + 8 * half) * NE_ + o0 + lh] = c[r] + bo;
  }
}

extern "C" void kernel_launch(void* const* d_in, const int* in_sizes, int n_in,
                              void* d_out, int out_size, void* d_ws, size_t ws_size,
                              hipStream_t stream) {
  (void)in_sizes; (void)n_in; (void)out_size; (void)d_ws; (void)ws_size;
  const float* query    = (const float*)d_in[0];
  const float* context  = (const float*)d_in[1];
  const float* memv     = (const float*)d_in[2];
  const float* w_logit  = (const float*)d_in[3];
  const float* b_logit  = (const float*)d_in[4];
  const float* temp     = (const float*)d_in[5];
  const float* w_reduce = (const float*)d_in[6];
  const float* b_reduce = (const float*)d_in[7];
  float* out = (float*)d_out;

  dim3 grid(N_ * (NQ_ / QT));   // 64 blocks
  dim3 block(512);              // 16 wave32s
  hipLaunchKernelGGL(attn_additive_fused, grid, block, 0, stream,
                     query, context, memv, w_logit, b_logit, temp,
                     w_reduce, b_reduce, out);
}